// Memory_73701638800014
// MI455X (gfx1250) — compile-verified
//
#include <hip/hip_runtime.h>
#include <math.h>

// CDNA5 / gfx1250, wave32. fp32 WMMA: D(16x16) = A(16x4) * B(4x16) + C.
typedef __attribute__((ext_vector_type(2))) float v2f;
typedef __attribute__((ext_vector_type(8))) float v8f;

#define HDIM 128
#define SLOTS 8
#define TB 8            // batches per block
#define AROWS 80        // 64 kv rows + 8 q rows + 8 zero pad (5 row tiles of 16)
#define PAD 132         // padded LDS row stride (floats) -> conflict-free frags

// Async global->LDS copy (CDNA5): per-lane 16B, tracked by ASYNCcnt.
// VDST = LDS byte offset (low 32 bits of a generic LDS pointer), VADDR = 64-bit global.
__device__ __forceinline__ void async_g2l_b128(unsigned lds_off, const void* gaddr) {
    asm volatile("global_load_async_to_lds_b128 %0, %1, off"
                 :: "v"(lds_off), "v"(gaddr) : "memory");
}
__device__ __forceinline__ void wait_asynccnt0() {
    asm volatile("s_wait_asynccnt 0x0" ::: "memory");
}
#define LDS_OFF(p) ((unsigned)(uintptr_t)(p))

// ---------------- prep: new_slot = sel*(1-gate) + o_emb_w*gate ----------------
__global__ void mem_prep_kernel(const float* __restrict__ memory,
                                const float* __restrict__ o_emb_w,
                                const float* __restrict__ forget_w,
                                const int* __restrict__ o_rg_p,
                                float* __restrict__ new_slot) {
    const int o_rg = o_rg_p[0];
    const int slot = threadIdx.x >> 5;   // 8 waves, one slot each
    const int lane = threadIdx.x & 31;
    const float* sel = memory + ((size_t)o_rg * SLOTS + slot) * HDIM;
    float acc = 0.f;
    for (int j = lane; j < HDIM; j += 32) {
        acc += o_emb_w[j] * forget_w[j];          // <o_emb, fw_lo>
        acc += sel[j]     * forget_w[HDIM + j];   // <sel_s, fw_hi>
    }
    for (int m = 16; m >= 1; m >>= 1) acc += __shfl_xor(acc, m, 32);
    const float g = 1.f / (1.f + expf(-acc));
    for (int j = lane; j < HDIM; j += 32)
        new_slot[slot * HDIM + j] = sel[j] * (1.f - g) + o_emb_w[j] * g;
}

// ---------------- main: async gather, dual GEMM via WMMA, softmax-attend ----------------
__global__ void mem_attn_kernel(const float* __restrict__ memory,
                                const float* __restrict__ o_emb_r,
                                const float* __restrict__ attn_W,
                                const float* __restrict__ sim_w,
                                const float* __restrict__ sim_b,
                                const int* __restrict__ o_rg_p,
                                const int* __restrict__ d_rg,
                                const float* __restrict__ new_slot,
                                float* __restrict__ out) {
    extern __shared__ float smem[];
    float* sW    = smem;                    // [128][PAD]  attn_W staged
    float* sA    = sW + HDIM * PAD;         // [AROWS][PAD] km rows + q rows
    float* sO    = sA + AROWS * PAD;        // [AROWS][PAD] kv / q GEMM results
    float* sDots = sO + AROWS * PAD;        // [AROWS] per-row sim_w dots
    float* sScore = sDots + AROWS;          // [TB*SLOTS] softmax scores

    const int tid   = threadIdx.x;
    const int block = blockIdx.x;
    const int o_rg  = o_rg_p[0];
    const unsigned sW_off = LDS_OFF(sW);
    const unsigned sA_off = LDS_OFF(sA);

    // ---- stage attn_W (128x128) into LDS via async DMA, padded ----
    #pragma unroll
    for (int j = 0; j < 16; ++j) {
        const int idx = tid + j * 256;           // 0..4095 float4 slots
        const int k = idx >> 5, n4 = idx & 31;
        async_g2l_b128(sW_off + (unsigned)(k * PAD + n4 * 4) * 4u,
                       attn_W + k * HDIM + n4 * 4);
    }
    // ---- stage A tile: rows 0..63 = km[b_local][s], 64..71 = o_emb_r, 72..79 = 0 ----
    {
        const int rowgrp = tid >> 5;             // 8 rows in flight
        const int c4 = tid & 31;                 // float4 column
        for (int r = 0; r < AROWS / 8; ++r) {
            const int row = rowgrp + r * 8;
            if (row < 64 + TB) {
                const float* src;
                if (row < 64) {
                    const int bl = row >> 3, s = row & 7;
                    const int idx = d_rg[block * TB + bl];
                    src = (idx == o_rg) ? (new_slot + s * HDIM)
                                        : (memory + ((size_t)idx * SLOTS + s) * HDIM);
                } else {
                    src = o_emb_r + (size_t)(block * TB + (row - 64)) * HDIM;
                }
                async_g2l_b128(sA_off + (unsigned)(row * PAD + c4 * 4) * 4u,
                               src + c4 * 4);
            } else {
                *(float4*)(sA + row * PAD + c4 * 4) = make_float4(0.f, 0.f, 0.f, 0.f);
            }
        }
    }
    wait_asynccnt0();
    __syncthreads();

    // ---- GEMM: wave w owns column tile nt=w; row tiles mt=0..4; K = 32 steps of 4 ----
    const int wave  = tid >> 5;
    const int lane  = tid & 31;
    const int hl    = lane & 15;                 // M index (A) / N index (B)
    const int khalf = (lane >> 4) << 1;          // lanes 16..31 hold K+2,K+3
    const int nt    = wave;

    v8f acc[5] = {};
    const float* bbase = sW + nt * 16 + hl;

    // double-buffered fragments: loads for step k+1 fly under WMMAs of step k
    v2f bf[2];
    v2f af[2][5];
    auto load_frags = [&](int k0, int buf) {
        bf[buf].x = bbase[(k0 + khalf) * PAD];
        bf[buf].y = bbase[(k0 + khalf + 1) * PAD];
        #pragma unroll
        for (int mt = 0; mt < 5; ++mt) {
            const float* ap = sA + (mt * 16 + hl) * PAD + k0 + khalf;
            af[buf][mt].x = ap[0];
            af[buf][mt].y = ap[1];
        }
    };
    load_frags(0, 0);
    #pragma unroll
    for (int k0 = 0; k0 < HDIM; k0 += 4) {
        const int cur = (k0 >> 2) & 1;
        if (k0 + 4 < HDIM) load_frags(k0 + 4, cur ^ 1);
        #pragma unroll
        for (int mt = 0; mt < 5; ++mt)
            acc[mt] = __builtin_amdgcn_wmma_f32_16x16x4_f32(
                false, af[cur][mt], false, bf[cur], (short)0, acc[mt], false, false);
    }

    // C/D layout: VGPR r -> row mt*16 + r + 8*(lane>=16), col nt*16 + (lane&15)
    #pragma unroll
    for (int mt = 0; mt < 5; ++mt) {
        const int rbase = mt * 16 + ((lane >> 4) << 3);
        #pragma unroll
        for (int r = 0; r < 8; ++r)
            sO[(rbase + r) * PAD + nt * 16 + hl] = acc[mt][r];
    }
    __syncthreads();

    // ---- per-row dots with sim_w halves (rows 0..63: sw_hi, 64..71: sw_lo) ----
    for (int i = 0; i < 9; ++i) {
        const int row = wave * 9 + i;            // waves cover rows 0..71
        const float* wv = (row < 64) ? (sim_w + HDIM) : sim_w;
        float s = 0.f;
        for (int j = lane; j < HDIM; j += 32)
            s += sO[row * PAD + j] * wv[j];
        for (int m = 16; m >= 1; m >>= 1) s += __shfl_xor(s, m, 32);
        if (lane == 0) sDots[row] = s;
    }
    __syncthreads();

    // ---- relu + softmax over 8 slots (one thread per batch) ----
    if (tid < TB) {
        const float lq = sDots[64 + tid];
        const float sb = sim_b[0];
        float lg[SLOTS];
        float mx = 0.f;                          // relu() >= 0 so 0 is a safe floor
        #pragma unroll
        for (int s = 0; s < SLOTS; ++s) {
            float v = lq + sDots[tid * SLOTS + s] + sb;
            v = v > 0.f ? v : 0.f;
            lg[s] = v;
            mx = v > mx ? v : mx;
        }
        float sum = 0.f;
        #pragma unroll
        for (int s = 0; s < SLOTS; ++s) { lg[s] = expf(lg[s] - mx); sum += lg[s]; }
        const float inv = 1.f / sum;
        #pragma unroll
        for (int s = 0; s < SLOTS; ++s) sScore[tid * SLOTS + s] = lg[s] * inv;
    }
    __syncthreads();

    // ---- weighted sum: out[b] = sum_s score[b,s] * kv[b,s,:] ----
    #pragma unroll
    for (int j = 0; j < 4; ++j) {
        const int idx = tid + j * 256;           // TB*HDIM = 1024 values
        const int b = idx >> 7, n = idx & 127;
        float o = 0.f;
        #pragma unroll
        for (int s = 0; s < SLOTS; ++s)
            o += sScore[b * SLOTS + s] * sO[(b * SLOTS + s) * PAD + n];
        out[((size_t)(block * TB) + b) * HDIM + n] = o;
    }
}

extern "C" void kernel_launch(void* const* d_in, const int* in_sizes, int n_in,
                              void* d_out, int out_size, void* d_ws, size_t ws_size,
                              hipStream_t stream) {
    (void)n_in; (void)out_size; (void)ws_size;
    const float* memory   = (const float*)d_in[0];
    const float* o_emb_w  = (const float*)d_in[1];
    const float* o_emb_r  = (const float*)d_in[2];
    const float* attn_W   = (const float*)d_in[3];
    const float* sim_w    = (const float*)d_in[4];
    const float* sim_b    = (const float*)d_in[5];
    const float* forget_w = (const float*)d_in[6];
    const int*   o_rg     = (const int*)d_in[7];
    const int*   d_rg     = (const int*)d_in[8];
    float* out      = (float*)d_out;
    float* new_slot = (float*)d_ws;              // 8*128 floats scratch

    mem_prep_kernel<<<1, 256, 0, stream>>>(memory, o_emb_w, forget_w, o_rg, new_slot);

    const int B = in_sizes[8];                   // 4096
    const size_t shmem =
        ((size_t)(HDIM + 2 * AROWS) * PAD + AROWS + TB * SLOTS) * sizeof(float);
    mem_attn_kernel<<<B / TB, 256, shmem, stream>>>(
        memory, o_emb_r, attn_W, sim_w, sim_b, o_rg, d_rg, new_slot, out);
}